// GraphSAGEInteractions_80788334838319
// MI455X (gfx1250) — compile-verified
//
#include <hip/hip_runtime.h>

#define HDIM 128
#define BN_EPS 1e-5f
#define NEG_INF_ENC 0x007FFFFFu   // encode(-inf): init value for segment-max

typedef __attribute__((ext_vector_type(2))) float v2f;
typedef __attribute__((ext_vector_type(8))) float v8f;

// Monotone order-preserving float<->uint encoding so atomicMax(u32) == float max.
__device__ __forceinline__ unsigned enc_f32(float f) {
    unsigned u = __float_as_uint(f);
    return (u & 0x80000000u) ? ~u : (u | 0x80000000u);
}
__device__ __forceinline__ float dec_f32(unsigned u) {
    return __uint_as_float((u & 0x80000000u) ? (u ^ 0x80000000u) : ~u);
}

// ---------------------------------------------------------------------------
// GEMM: out[n][o] = relu?( A[n][:] . W[o][:] + bias[o] ), K = 128, 128 cols.
// One wave per 16x16 output tile, V_WMMA_F32_16X16X4_F32, K-loop of 32 WMMAs.
// ---------------------------------------------------------------------------
__global__ void gemm128_bias(const float* __restrict__ A,
                             const float* __restrict__ W,
                             const float* __restrict__ bias,
                             float* __restrict__ out,
                             int nrows, int do_relu) {
    const int wave  = (int)((blockIdx.x * blockDim.x + threadIdx.x) >> 5);
    const int lane  = threadIdx.x & 31;
    const int row0  = (wave >> 3) << 4;      // 8 column tiles per row band
    const int col0  = (wave & 7) << 4;
    if (row0 >= nrows) return;

    const int t     = lane & 15;             // M index for A frag / N index for B,C,D
    const int khalf = lane >> 4;             // which K pair this half-wave owns

    int arow_i = row0 + t;                   // clamp for ragged tail (N%16, unused here)
    if (arow_i >= nrows) arow_i = nrows - 1;
    const float* arow = A + (size_t)arow_i * HDIM + khalf * 2;
    const float* wrow = W + (size_t)(col0 + t) * HDIM + khalf * 2;
    __builtin_prefetch(wrow, 0, 3);

    const float bcol = bias[col0 + t];
    v8f c;
#pragma unroll
    for (int r = 0; r < 8; ++r) c[r] = bcol;

#pragma unroll
    for (int k0 = 0; k0 < HDIM; k0 += 4) {
        v2f a = *(const v2f*)(arow + k0);
        v2f b = *(const v2f*)(wrow + k0);
        c = __builtin_amdgcn_wmma_f32_16x16x4_f32(false, a, false, b,
                                                  (short)0, c, false, false);
    }

#pragma unroll
    for (int r = 0; r < 8; ++r) {
        const int orow = row0 + khalf * 8 + r;
        if (orow < nrows) {
            float v = c[r];
            if (do_relu) v = fmaxf(v, 0.0f);
            out[(size_t)orow * HDIM + col0 + t] = v;
        }
    }
}

// ---------------------------------------------------------------------------
// SAGE layer: h = AGG@WLt + bias + X@WRt, then BatchNorm(eval) + ReLU.
// Two 32-WMMA K-loops chained into one accumulator per 16x16 tile.
// ---------------------------------------------------------------------------
__global__ void sage_layer_gemm(const float* __restrict__ AGG,
                                const float* __restrict__ WL,
                                const float* __restrict__ X,
                                const float* __restrict__ WR,
                                const float* __restrict__ bias,
                                const float* __restrict__ bn_g,
                                const float* __restrict__ bn_b,
                                const float* __restrict__ bn_m,
                                const float* __restrict__ bn_v,
                                float* __restrict__ out, int nrows) {
    const int wave  = (int)((blockIdx.x * blockDim.x + threadIdx.x) >> 5);
    const int lane  = threadIdx.x & 31;
    const int row0  = (wave >> 3) << 4;
    const int col0  = (wave & 7) << 4;
    if (row0 >= nrows) return;

    const int t     = lane & 15;
    const int khalf = lane >> 4;
    const int col   = col0 + t;

    int arow_i = row0 + t;
    if (arow_i >= nrows) arow_i = nrows - 1;
    const float* aggrow = AGG + (size_t)arow_i * HDIM + khalf * 2;
    const float* xrow   = X   + (size_t)arow_i * HDIM + khalf * 2;
    const float* wlrow  = WL  + (size_t)col * HDIM + khalf * 2;
    const float* wrrow  = WR  + (size_t)col * HDIM + khalf * 2;
    __builtin_prefetch(wlrow, 0, 3);
    __builtin_prefetch(wrrow, 0, 3);

    const float bcol = bias[col];
    v8f c;
#pragma unroll
    for (int r = 0; r < 8; ++r) c[r] = bcol;

#pragma unroll
    for (int k0 = 0; k0 < HDIM; k0 += 4) {          // lin_l on aggregated neighbors
        v2f a = *(const v2f*)(aggrow + k0);
        v2f b = *(const v2f*)(wlrow + k0);
        c = __builtin_amdgcn_wmma_f32_16x16x4_f32(false, a, false, b,
                                                  (short)0, c, false, false);
    }
#pragma unroll
    for (int k0 = 0; k0 < HDIM; k0 += 4) {          // lin_r on root features
        v2f a = *(const v2f*)(xrow + k0);
        v2f b = *(const v2f*)(wrrow + k0);
        c = __builtin_amdgcn_wmma_f32_16x16x4_f32(false, a, false, b,
                                                  (short)0, c, false, false);
    }

    // Fused BatchNorm1d (running stats) + ReLU epilogue; column fixed per lane.
    const float s  = bn_g[col] * rsqrtf(bn_v[col] + BN_EPS);
    const float mu = bn_m[col];
    const float bb = bn_b[col];
#pragma unroll
    for (int r = 0; r < 8; ++r) {
        const int orow = row0 + khalf * 8 + r;
        if (orow < nrows) {
            float v = (c[r] - mu) * s + bb;
            out[(size_t)orow * HDIM + col0 + t] = fmaxf(v, 0.0f);
        }
    }
}

// ---------------------------------------------------------------------------
// Scatter-max over edges: one wave per edge, float4 per lane, u32 atomicMax.
// ---------------------------------------------------------------------------
__global__ void init_agg(unsigned* __restrict__ agg, long long count) {
    long long i = (long long)blockIdx.x * blockDim.x + threadIdx.x;
    if (i < count) agg[i] = NEG_INF_ENC;
}

__global__ void scatter_max_edges(const int* __restrict__ ei,
                                  const float* __restrict__ X,
                                  unsigned* __restrict__ agg, int E) {
    long long tid = (long long)blockIdx.x * blockDim.x + threadIdx.x;
    int e = (int)(tid >> 5);
    if (e >= E) return;
    const int lane = (int)(tid & 31);
    const int dst = ei[e];        // edge_index[0] = destination
    const int src = ei[E + e];    // edge_index[1] = source (message origin)
    const float4 v = *(const float4*)(X + (size_t)src * HDIM + lane * 4);
    unsigned* a = agg + (size_t)dst * HDIM + lane * 4;
    atomicMax(a + 0, enc_f32(v.x));
    atomicMax(a + 1, enc_f32(v.y));
    atomicMax(a + 2, enc_f32(v.z));
    atomicMax(a + 3, enc_f32(v.w));
}

__global__ void decode_agg(const unsigned* __restrict__ agg,
                           float* __restrict__ out, long long count) {
    long long i = (long long)blockIdx.x * blockDim.x + threadIdx.x;
    if (i < count) {
        unsigned u = agg[i];
        out[i] = (u == NEG_INF_ENC) ? 0.0f : dec_f32(u);   // isolated nodes -> 0
    }
}

// ---------------------------------------------------------------------------
// fc2 head: out[n][0..3] = Y[n][:] . W2[j][:] + b2[j]; wave per row, shfl reduce.
// ---------------------------------------------------------------------------
__global__ void fc2_head(const float* __restrict__ Y,
                         const float* __restrict__ W2,
                         const float* __restrict__ B2,
                         float* __restrict__ out, int nrows) {
    const int wave = (int)((blockIdx.x * blockDim.x + threadIdx.x) >> 5);
    const int lane = threadIdx.x & 31;
    if (wave >= nrows) return;

    const float4 y = *(const float4*)(Y + (size_t)wave * HDIM + lane * 4);
    float acc[4];
#pragma unroll
    for (int j = 0; j < 4; ++j) {
        const float4 w = *(const float4*)(W2 + (size_t)j * HDIM + lane * 4);
        acc[j] = y.x * w.x + y.y * w.y + y.z * w.z + y.w * w.w;
    }
#pragma unroll
    for (int off = 16; off > 0; off >>= 1) {
#pragma unroll
        for (int j = 0; j < 4; ++j) acc[j] += __shfl_xor(acc[j], off, 32);
    }
    if (lane == 0) {
#pragma unroll
        for (int j = 0; j < 4; ++j) out[(size_t)wave * 4 + j] = acc[j] + B2[j];
    }
}

// ---------------------------------------------------------------------------
extern "C" void kernel_launch(void* const* d_in, const int* in_sizes, int n_in,
                              void* d_out, int out_size, void* d_ws, size_t ws_size,
                              hipStream_t stream) {
    const float* features = (const float*)d_in[0];
    const int*   ei       = (const int*)d_in[1];
    const float* w_proj = (const float*)d_in[2];
    const float* b_proj = (const float*)d_in[3];
    const float* w_l0 = (const float*)d_in[4];
    const float* b_l0 = (const float*)d_in[5];
    const float* w_r0 = (const float*)d_in[6];
    const float* bn_g0 = (const float*)d_in[7];
    const float* bn_b0 = (const float*)d_in[8];
    const float* bn_m0 = (const float*)d_in[9];
    const float* bn_v0 = (const float*)d_in[10];
    const float* w_l1 = (const float*)d_in[11];
    const float* b_l1 = (const float*)d_in[12];
    const float* w_r1 = (const float*)d_in[13];
    const float* bn_g1 = (const float*)d_in[14];
    const float* bn_b1 = (const float*)d_in[15];
    const float* bn_m1 = (const float*)d_in[16];
    const float* bn_v1 = (const float*)d_in[17];
    const float* w_fc1 = (const float*)d_in[18];
    const float* b_fc1 = (const float*)d_in[19];
    const float* w_fc2 = (const float*)d_in[20];
    const float* b_fc2 = (const float*)d_in[21];
    float* outp = (float*)d_out;

    const int N = in_sizes[0] / HDIM;
    const int E = in_sizes[1] / 2;
    const long long NH = (long long)N * HDIM;

    const size_t SZ = (size_t)N * HDIM * sizeof(float);
    float*    X    = (float*)d_ws;                        // current node features
    unsigned* AGG  = (unsigned*)((char*)d_ws + SZ);       // encoded scatter-max
    float*    AGGF = (float*)((char*)d_ws + 2 * SZ);      // decoded aggregation
    float*    H1   = (float*)((char*)d_ws + 3 * SZ);      // ping-pong buffer

    const dim3 blk(128);                                  // 4 waves per block
    const int row_tiles   = (N + 15) / 16;
    const int gemm_blocks = (row_tiles * 8 + 3) / 4;      // 8 col-tiles, /4 waves
    const int ew_blocks   = (int)((NH + 255) / 256);
    const long long sc_threads = (long long)E * 32;
    const int sc_blocks   = (int)((sc_threads + 255) / 256);
    const int fc2_blocks  = (N + 7) / 8;                  // 8 waves (rows) per block

    // 1) projection: X = features @ w_proj^T + b_proj
    gemm128_bias<<<gemm_blocks, blk, 0, stream>>>(features, w_proj, b_proj, X, N, 0);

    // 2) SAGE layer 0: X -> H1
    init_agg<<<ew_blocks, 256, 0, stream>>>(AGG, NH);
    scatter_max_edges<<<sc_blocks, 256, 0, stream>>>(ei, X, AGG, E);
    decode_agg<<<ew_blocks, 256, 0, stream>>>(AGG, AGGF, NH);
    sage_layer_gemm<<<gemm_blocks, blk, 0, stream>>>(AGGF, w_l0, X, w_r0, b_l0,
                                                     bn_g0, bn_b0, bn_m0, bn_v0, H1, N);

    // 3) SAGE layer 1: H1 -> X
    init_agg<<<ew_blocks, 256, 0, stream>>>(AGG, NH);
    scatter_max_edges<<<sc_blocks, 256, 0, stream>>>(ei, H1, AGG, E);
    decode_agg<<<ew_blocks, 256, 0, stream>>>(AGG, AGGF, NH);
    sage_layer_gemm<<<gemm_blocks, blk, 0, stream>>>(AGGF, w_l1, H1, w_r1, b_l1,
                                                     bn_g1, bn_b1, bn_m1, bn_v1, X, N);

    // 4) fc1 + ReLU: X -> H1
    gemm128_bias<<<gemm_blocks, blk, 0, stream>>>(X, w_fc1, b_fc1, H1, N, 1);

    // 5) fc2: H1 -> out [N,4]
    fc2_head<<<fc2_blocks, 256, 0, stream>>>(H1, w_fc2, b_fc2, outp, N);
}